// _BRB_DCN_Module_39101382262996
// MI455X (gfx1250) — compile-verified
//
#include <hip/hip_runtime.h>

#define DIM   1024
#define KC    1024
#define CHUNK 512          // D staged in LDS per pass
#define APITCH 516         // LDS row pitch (floats): bank stride 4 -> conflict-free frag loads

typedef float v8f __attribute__((ext_vector_type(8)));
typedef float v2f __attribute__((ext_vector_type(2)));
typedef int   v4i __attribute__((ext_vector_type(4)));

#if defined(__has_builtin)
#if __has_builtin(__builtin_amdgcn_global_load_async_to_lds_b128) && \
    __has_builtin(__builtin_amdgcn_s_wait_asynccnt)
#define HAS_ASYNC_LDS 1
typedef __attribute__((address_space(1))) v4i* gp128;   // global b128 pointer
typedef __attribute__((address_space(3))) v4i* lp128;   // LDS b128 pointer
#endif
#endif

// ---------------- c2[k] = sum_d centers[k][d]^2 ----------------
__global__ __launch_bounds__(256) void k_c2(const float* __restrict__ C,
                                            float* __restrict__ c2) {
    const int k = blockIdx.x;
    const int t = threadIdx.x;
    const float4 v = ((const float4*)(C + (size_t)k * DIM))[t]; // 256*4 = 1024
    float p = v.x * v.x + v.y * v.y + v.z * v.z + v.w * v.w;
    for (int off = 16; off > 0; off >>= 1) p += __shfl_down(p, off, 32);
    __shared__ float ws[8];
    if ((t & 31) == 0) ws[t >> 5] = p;
    __syncthreads();
    if (t == 0) {
        float s = 0.f;
        #pragma unroll
        for (int i = 0; i < 8; ++i) s += ws[i];
        c2[k] = s;
    }
}

// ---------------- zero sums (in d_out) + float counts ----------------
__global__ void k_zero(float* __restrict__ sums, float* __restrict__ cnt, int n) {
    int i = blockIdx.x * 256 + threadIdx.x;
    if (i < n) sums[i] = 0.f;
    if (i < KC) cnt[i] = 0.f;
}

// ---------------- fused dot (WMMA f32) + argmin -> labels ----------------
// Block: 256 thr = 8 waves. Block owns 16 rows of E; wave w owns cols [w*128, w*128+128).
// score[n][k] = c2[k] - 2 * dot(E[n], C[k]);  label[n] = argmin_k score
__global__ __launch_bounds__(256) void k_argmin(const float* __restrict__ E,
                                                const float* __restrict__ C,
                                                const float* __restrict__ c2,
                                                int* __restrict__ labels) {
    __shared__ float smA[16 * APITCH];        // 16 rows x CHUNK, pitch 516 -> ~33 KB
    __shared__ float redV[16 * 8];            // [row][wave]
    __shared__ int   redI[16 * 8];

    const int tid     = threadIdx.x;
    const int rowbase = blockIdx.x * 16;
    const int lane    = tid & 31;
    const int wid     = tid >> 5;
    const int half    = lane >> 4;            // C/D half: rows 0-7 vs 8-15
    const int l16     = lane & 15;
    const int koff    = half * 2;             // A/B K sub-offset per ISA layout
    const int colbase = wid * 128;

    v8f acc[8];
    #pragma unroll
    for (int t = 0; t < 8; ++t) acc[t] = (v8f)(0.0f);

    // per-lane global base pointers for the 8 B tiles (centers rows = columns)
    const float* brow[8];
    #pragma unroll
    for (int t = 0; t < 8; ++t)
        brow[t] = C + (size_t)(colbase + t * 16 + l16) * DIM + koff;

    const float* arow = &smA[l16 * APITCH + koff];

    for (int ch = 0; ch < DIM / CHUNK; ++ch) {
        // stage 16 x CHUNK slice of E into LDS (coalesced 16B granules)
        const float4* E4 = (const float4*)(E + (size_t)rowbase * DIM + ch * CHUNK);
#if defined(HAS_ASYNC_LDS)
        #pragma unroll
        for (int j = 0; j < (16 * CHUNK / 4) / 256; ++j) {   // 8 iters
            int idx = j * 256 + tid;
            int r   = idx >> 7;          // / (CHUNK/4)
            int c4  = idx & 127;
            __builtin_amdgcn_global_load_async_to_lds_b128(
                (gp128)(E4 + (size_t)r * (DIM / 4) + c4),
                (lp128)&smA[r * APITCH + c4 * 4],
                0, 0);
        }
        __builtin_amdgcn_s_wait_asynccnt(0);
#else
        #pragma unroll
        for (int j = 0; j < (16 * CHUNK / 4) / 256; ++j) {   // 8 iters
            int idx = j * 256 + tid;
            int r   = idx >> 7;          // / (CHUNK/4)
            int c4  = idx & 127;
            float4 v = E4[(size_t)r * (DIM / 4) + c4];
            *(float4*)&smA[r * APITCH + c4 * 4] = v;
        }
#endif
        __syncthreads();

        #pragma unroll 2
        for (int dl = 0; dl < CHUNK; dl += 4) {
            v2f av = *(const v2f*)(arow + dl);
            #pragma unroll
            for (int t = 0; t < 8; ++t) {
                v2f bv = *(const v2f*)(brow[t] + ch * CHUNK + dl);
                acc[t] = __builtin_amdgcn_wmma_f32_16x16x4_f32(
                    false, av, false, bv, (short)0, acc[t], false, false);
            }
        }
        __syncthreads();
    }

    // ---- argmin: cross-lane (16-lane halves match C-layout) then cross-wave ----
    float bestV[8];
    int   bestI[8];
    #pragma unroll
    for (int r = 0; r < 8; ++r) { bestV[r] = 3.402823466e38f; bestI[r] = 0x7fffffff; }

    #pragma unroll
    for (int t = 0; t < 8; ++t) {
        const int col = colbase + t * 16 + l16;
        const float cc = c2[col];
        #pragma unroll
        for (int r = 0; r < 8; ++r) {
            float v = cc - 2.0f * acc[t][r];
            int   i = col;
            #pragma unroll
            for (int m = 1; m <= 8; m <<= 1) {        // stays within 16-lane half
                float ov = __shfl_xor(v, m, 32);
                int   oi = __shfl_xor(i, m, 32);
                if (ov < v || (ov == v && oi < i)) { v = ov; i = oi; }
            }
            if (v < bestV[r] || (v == bestV[r] && i < bestI[r])) { bestV[r] = v; bestI[r] = i; }
        }
    }
    if (l16 == 0) {                                   // lane 0 (rows 0-7), lane 16 (rows 8-15)
        #pragma unroll
        for (int r = 0; r < 8; ++r) {
            int row = half * 8 + r;
            redV[row * 8 + wid] = bestV[r];
            redI[row * 8 + wid] = bestI[r];
        }
    }
    __syncthreads();
    if (tid < 16) {
        float bv = redV[tid * 8];
        int   bi = redI[tid * 8];
        #pragma unroll
        for (int w = 1; w < 8; ++w) {
            float v = redV[tid * 8 + w];
            int   i = redI[tid * 8 + w];
            if (v < bv || (v == bv && i < bi)) { bv = v; bi = i; }
        }
        labels[rowbase + tid] = bi;
    }
}

// ---------------- segmented sum via hardware f32 atomics ----------------
__global__ __launch_bounds__(256) void k_scatter(const float* __restrict__ E,
                                                 const int* __restrict__ labels,
                                                 float* __restrict__ sums,
                                                 float* __restrict__ cnt) {
    const int i = blockIdx.x;
    const int t = threadIdx.x;
    const int lab = labels[i];
    const float4 v = ((const float4*)(E + (size_t)i * DIM))[t];
    float* dst = sums + (size_t)lab * DIM + t * 4;
    unsafeAtomicAdd(dst + 0, v.x);
    unsafeAtomicAdd(dst + 1, v.y);
    unsafeAtomicAdd(dst + 2, v.z);
    unsafeAtomicAdd(dst + 3, v.w);
    if (t == 0) unsafeAtomicAdd(&cnt[lab], 1.0f);
}

// ---------------- EMA finalize, in place on d_out (holds sums) ----------------
__global__ void k_final(const float* __restrict__ centers, const int* __restrict__ count,
                        const float* __restrict__ cnt, float* __restrict__ out, int n) {
    int idx = blockIdx.x * 256 + threadIdx.x;
    if (idx >= n) return;
    int k = idx >> 10;                     // / DIM
    float ms = cnt[k];
    float c  = centers[idx];
    float r  = c;
    if (ms > 0.f) {
        float ccn = 0.5f * (float)count[k] + 0.5f * ms;   // CENTER_LR = 0.5
        float lr  = 1.0f / (ccn + 1.0f);
        float bc  = out[idx] / ms;         // sums accumulated into d_out
        r = c + lr * (bc - c);             // (1-lr)*c + lr*bc
    }
    out[idx] = r;
}

extern "C" void kernel_launch(void* const* d_in, const int* in_sizes, int n_in,
                              void* d_out, int out_size, void* d_ws, size_t ws_size,
                              hipStream_t stream) {
    const float* E     = (const float*)d_in[0];
    const float* C     = (const float*)d_in[1];
    const int*   count = (const int*)d_in[2];
    float* out = (float*)d_out;

    const int N = in_sizes[0] / DIM;       // 16384

    char* ws   = (char*)d_ws;
    int*   labels = (int*)ws;                                        // N ints
    float* cnt    = (float*)(ws + (size_t)N * sizeof(int));          // KC floats
    float* c2     = (float*)(ws + (size_t)N * sizeof(int) + KC * sizeof(float));

    k_zero<<<(KC * DIM + 255) / 256, 256, 0, stream>>>(out, cnt, KC * DIM);
    k_c2<<<KC, 256, 0, stream>>>(C, c2);
    k_argmin<<<N / 16, 256, 0, stream>>>(E, C, c2, labels);
    k_scatter<<<N, 256, 0, stream>>>(E, labels, out, cnt);
    k_final<<<(KC * DIM + 255) / 256, 256, 0, stream>>>(C, count, cnt, out, KC * DIM);
}